// GMMLoss2_pca_73547019977337
// MI455X (gfx1250) — compile-verified
//
#include <hip/hip_runtime.h>
#include <math.h>

#define B_ROWS 8192
#define D_DIM  1024
#define KD     128
#define L_LAB  10
#define POWER_ITERS 7

typedef float v2f __attribute__((ext_vector_type(2)));
typedef float v8f __attribute__((ext_vector_type(8)));
typedef int   v4i __attribute__((ext_vector_type(4)));

__device__ __forceinline__ v8f wmma_f32x4(v2f a, v2f b, v8f c) {
  return __builtin_amdgcn_wmma_f32_16x16x4_f32(false, a, false, b, (short)0, c,
                                               false, false);
}

// ---------------- CDNA5 async global->LDS copy (ASYNCcnt path) -----------
#if defined(__has_builtin)
#if __has_builtin(__builtin_amdgcn_global_load_async_to_lds_b128)
#define HAVE_ASYNC 1
#endif
#endif
#ifndef HAVE_ASYNC
#define HAVE_ASYNC 0
#endif

#if HAVE_ASYNC
// Builtin signature (from clang diagnostic): param0 = v4i in global AS
// (printed as "int __vector(4) __device__ *", non-const), param1 = LDS side,
// then imm offset, imm cpol.
typedef v4i __attribute__((address_space(1)))* v4i_gptr;
typedef v4i __attribute__((address_space(3)))* v4i_lptr;

__device__ __forceinline__ void gld_async_b128(void* lds, const void* g) {
  __builtin_amdgcn_global_load_async_to_lds_b128((v4i_gptr)g, (v4i_lptr)lds,
                                                 0, 0);
}
#endif

__device__ __forceinline__ void wait_async0() {
#if defined(__has_builtin) && __has_builtin(__builtin_amdgcn_s_wait_asynccnt)
  __builtin_amdgcn_s_wait_asynccnt(0);
#else
  asm volatile("s_wait_asynccnt 0" ::: "memory");
#endif
}

// ---------------- wave-level 32x32 WMMA micro-kernels --------------------
// sA rows padded to 136, sB rows to 72: the two lane-halves read LDS rows
// k0 and k0+2; 2*136 = 272 = 16 mod 64 banks and 2*72 = 144 = 16 mod 64, so
// the two 16-lane bank ranges are disjoint (wm/wn multiples of 16).
__device__ __forceinline__ void gram_compute(const float (*sA)[136],
                                             const float (*sB)[72],
                                             v8f acc[2][2], int wm, int wn,
                                             int half, int lm) {
#pragma unroll
  for (int kk = 0; kk < 8; ++kk) {
    int k0 = kk * 4 + 2 * half;
#pragma unroll
    for (int ti = 0; ti < 2; ++ti) {
      v2f a;
      a.x = sA[k0][wm + ti * 16 + lm];
      a.y = sA[k0 + 1][wm + ti * 16 + lm];
#pragma unroll
      for (int tj = 0; tj < 2; ++tj) {
        v2f b;
        b.x = sB[k0][wn + tj * 16 + lm];
        b.y = sB[k0 + 1][wn + tj * 16 + lm];
        acc[ti][tj] = wmma_f32x4(a, b, acc[ti][tj]);
      }
    }
  }
}

// A staged transposed (sAT[m][k], pad 36): a.x/a.y contiguous -> ds_load_b64,
// lane stride 36 dwords -> conflict-free (gcd(36,64)=4, 16 lanes distinct;
// halves offset by 2 dwords -> disjoint bank parity).
__device__ __forceinline__ void gemm_compute(const float (*sAT)[36],
                                             const float (*sB)[72],
                                             v8f acc[2][2], int wm, int wn,
                                             int half, int lm) {
#pragma unroll
  for (int kk = 0; kk < 8; ++kk) {
    int k0 = kk * 4 + 2 * half;
#pragma unroll
    for (int ti = 0; ti < 2; ++ti) {
      v2f a;
      a.x = sAT[wm + ti * 16 + lm][k0];
      a.y = sAT[wm + ti * 16 + lm][k0 + 1];
#pragma unroll
      for (int tj = 0; tj < 2; ++tj) {
        v2f b;
        b.x = sB[k0][wn + tj * 16 + lm];
        b.y = sB[k0 + 1][wn + tj * 16 + lm];
        acc[ti][tj] = wmma_f32x4(a, b, acc[ti][tj]);
      }
    }
  }
}

// -------------------------------------------------------------------------
// out(+z*ncols^2) = (mask_z X)^T * X, raw (uncentered) Gram.  X row-major
// [nrows x ncols].  Block tile 128(M) x 64(N), 8 waves of 32x32 WMMA tiles.
// labels==nullptr: pure copy staging -> async b128 double-buffered DMA.
// labels!=nullptr: A-side rows zeroed where labels[row] != blockIdx.z.
// -------------------------------------------------------------------------
__global__ __launch_bounds__(256) void gram_kernel(
    const float* __restrict__ X, const int* __restrict__ labels,
    float* __restrict__ out, int ncols, int nrows) {
  __shared__ float sA[2][32][136];
  __shared__ float sB[2][32][72];
  const int t = threadIdx.x;
  const int Mb = blockIdx.x * 128;
  const int Nb = blockIdx.y * 64;
  const int target = blockIdx.z;
  float* outp = out + (size_t)blockIdx.z * (size_t)ncols * (size_t)ncols;

  const int wave = t >> 5, lane = t & 31;
  const int wm = (wave & 3) * 32, wn = (wave >> 2) * 32;
  const int half = lane >> 4, lm = lane & 15;

  v8f acc[2][2];
  const v8f zv = {0.f, 0.f, 0.f, 0.f, 0.f, 0.f, 0.f, 0.f};
#pragma unroll
  for (int i = 0; i < 2; ++i)
#pragma unroll
    for (int j = 0; j < 2; ++j) acc[i][j] = zv;

  if (labels == nullptr) {
#if HAVE_ASYNC
    auto stage = [&](int buf, int kb) {
#pragma unroll
      for (int i = 0; i < 4; ++i) {  // A panel: 32x128 = 1024 x 16B chunks
        int e = t + i * 256;
        int k = e >> 5, mc = (e & 31) * 4;
        gld_async_b128(&sA[buf][k][mc],
                       &X[(size_t)(kb + k) * ncols + Mb + mc]);
      }
#pragma unroll
      for (int i = 0; i < 2; ++i) {  // B panel: 32x64 = 512 x 16B chunks
        int e = t + i * 256;
        int k = e >> 4, nc = (e & 15) * 4;
        gld_async_b128(&sB[buf][k][nc],
                       &X[(size_t)(kb + k) * ncols + Nb + nc]);
      }
    };
    stage(0, 0);
    int buf = 0;
    for (int kb = 0; kb < nrows; kb += 32) {
      wait_async0();       // own DMA done
      __syncthreads();     // everyone's DMA done / prev reads done
      if (kb + 32 < nrows) stage(buf ^ 1, kb + 32);
      gram_compute(sA[buf], sB[buf], acc, wm, wn, half, lm);
      buf ^= 1;
    }
#else
    for (int kb = 0; kb < nrows; kb += 32) {
      __syncthreads();
#pragma unroll
      for (int i = 0; i < 16; ++i) {
        int e = t + i * 256;
        int k = e >> 7, m = e & 127;
        sA[0][k][m] = X[(size_t)(kb + k) * ncols + Mb + m];
      }
#pragma unroll
      for (int i = 0; i < 8; ++i) {
        int e = t + i * 256;
        int k = e >> 6, n = e & 63;
        sB[0][k][n] = X[(size_t)(kb + k) * ncols + Nb + n];
      }
      __syncthreads();
      gram_compute(sA[0], sB[0], acc, wm, wn, half, lm);
    }
#endif
  } else {
    for (int kb = 0; kb < nrows; kb += 32) {
      __syncthreads();
#pragma unroll
      for (int i = 0; i < 16; ++i) {
        int e = t + i * 256;
        int k = e >> 7, m = e & 127;
        int row = kb + k;
        float v = X[(size_t)row * ncols + Mb + m];
        sA[0][k][m] = (labels[row] == target) ? v : 0.0f;
      }
#pragma unroll
      for (int i = 0; i < 8; ++i) {
        int e = t + i * 256;
        int k = e >> 6, n = e & 63;
        sB[0][k][n] = X[(size_t)(kb + k) * ncols + Nb + n];
      }
      if (kb + 32 < nrows)
        __builtin_prefetch(&X[(size_t)(kb + 32 + (t >> 3)) * ncols + Mb + (t & 7) * 16], 0, 0);
      __syncthreads();
      gram_compute(sA[0], sB[0], acc, wm, wn, half, lm);
    }
  }

#pragma unroll
  for (int ti = 0; ti < 2; ++ti)
#pragma unroll
    for (int tj = 0; tj < 2; ++tj)
#pragma unroll
      for (int v = 0; v < 8; ++v) {
        int row = Mb + wm + ti * 16 + half * 8 + v;
        int col = Nb + wn + tj * 16 + lm;
        outp[(size_t)row * ncols + col] = acc[ti][tj][v];
      }
}

// -------------------------------------------------------------------------
// out[M x 128] = A * Bm,  A row-major [M x Kdim] (lda), Bm row-major
// [Kdim x 128].  grid = (M/128, 2).  Async b128 double-buffered staging.
// -------------------------------------------------------------------------
__global__ __launch_bounds__(256) void gemm_kernel(
    const float* __restrict__ A, int lda, const float* __restrict__ Bm,
    float* __restrict__ out, int Kdim) {
  __shared__ float sAT[2][128][36];
  __shared__ float sB[2][32][72];
  const int t = threadIdx.x;
  const int Mb = blockIdx.x * 128;
  const int Nb = blockIdx.y * 64;

  const int wave = t >> 5, lane = t & 31;
  const int wm = (wave & 3) * 32, wn = (wave >> 2) * 32;
  const int half = lane >> 4, lm = lane & 15;

  v8f acc[2][2];
  const v8f zv = {0.f, 0.f, 0.f, 0.f, 0.f, 0.f, 0.f, 0.f};
#pragma unroll
  for (int i = 0; i < 2; ++i)
#pragma unroll
    for (int j = 0; j < 2; ++j) acc[i][j] = zv;

#if HAVE_ASYNC
  auto stage = [&](int buf, int kb) {
#pragma unroll
    for (int i = 0; i < 4; ++i) {  // A: 128 rows x 8 chunks (transposed copy)
      int e = t + i * 256;
      int m = e >> 3, kc = (e & 7) * 4;
      gld_async_b128(&sAT[buf][m][kc],
                     &A[(size_t)(Mb + m) * lda + kb + kc]);
    }
#pragma unroll
    for (int i = 0; i < 2; ++i) {  // B: 32x64
      int e = t + i * 256;
      int k = e >> 4, nc = (e & 15) * 4;
      gld_async_b128(&sB[buf][k][nc],
                     &Bm[(size_t)(kb + k) * 128 + Nb + nc]);
    }
  };
  stage(0, 0);
  int buf = 0;
  for (int kb = 0; kb < Kdim; kb += 32) {
    wait_async0();
    __syncthreads();
    if (kb + 32 < Kdim) stage(buf ^ 1, kb + 32);
    gemm_compute(sAT[buf], sB[buf], acc, wm, wn, half, lm);
    buf ^= 1;
  }
#else
  for (int kb = 0; kb < Kdim; kb += 32) {
    __syncthreads();
#pragma unroll
    for (int i = 0; i < 16; ++i) {
      int e = t + i * 256;
      int k = e & 31, m = e >> 5;
      sAT[0][m][k] = A[(size_t)(Mb + m) * lda + kb + k];
    }
#pragma unroll
    for (int i = 0; i < 8; ++i) {
      int e = t + i * 256;
      int k = e >> 6, n = e & 63;
      sB[0][k][n] = Bm[(size_t)(kb + k) * 128 + Nb + n];
    }
    __syncthreads();
    gemm_compute(sAT[0], sB[0], acc, wm, wn, half, lm);
  }
#endif

#pragma unroll
  for (int ti = 0; ti < 2; ++ti)
#pragma unroll
    for (int tj = 0; tj < 2; ++tj)
#pragma unroll
      for (int v = 0; v < 8; ++v) {
        int row = Mb + wm + ti * 16 + half * 8 + v;
        int col = Nb + wn + tj * 16 + lm;
        out[(size_t)row * 128 + col] = acc[ti][tj][v];
      }
}

// ------------------------- small helper kernels --------------------------
__global__ void colmean_kernel(const float* __restrict__ X, int rows, int cols,
                               float* __restrict__ out) {
  int c = blockIdx.x * blockDim.x + threadIdx.x;
  if (c >= cols) return;
  float s = 0.0f;
  for (int r = 0; r < rows; ++r) s += X[(size_t)r * cols + c];
  out[c] = s / (float)rows;
}

__global__ void init_kernel(float* __restrict__ U, int n) {
  int i = blockIdx.x * 256 + threadIdx.x;
  if (i < n) {
    unsigned h = (unsigned)i * 2654435761u;
    h ^= h >> 16; h *= 2246822519u; h ^= h >> 13;
    U[i] = ((float)(h & 0xFFFFu) / 65536.0f) - 0.5f;
  }
}

__global__ void zero_kernel(float* __restrict__ p, int n) {
  int i = blockIdx.x * 256 + threadIdx.x;
  if (i < n) p[i] = 0.0f;
}

// C -= B_ROWS * mean mean^T   (de-centering correction for x^T x)
__global__ void rank1_ep(float* __restrict__ C, const float* __restrict__ mean) {
  int idx = blockIdx.x * 256 + threadIdx.x;  // 1024*1024
  int i = idx >> 10, j = idx & 1023;
  C[idx] -= (float)B_ROWS * mean[i] * mean[j];
}

// w = mean^T * Uk  (row shift so that z = xc @ Uk)
__global__ void meanproj_kernel(const float* __restrict__ mean,
                                const float* __restrict__ Uk,
                                float* __restrict__ w) {
  int j = threadIdx.x;  // 128
  float s = 0.0f;
  for (int d = 0; d < D_DIM; ++d) s += mean[d] * Uk[d * KD + j];
  w[j] = s;
}

__global__ void zshift_kernel(float* __restrict__ z, const float* __restrict__ w) {
  int idx = blockIdx.x * 256 + threadIdx.x;  // B*KD
  z[idx] -= w[idx & 127];
}

__global__ void accum_kernel(const float* __restrict__ z,
                             const int* __restrict__ labels,
                             float* __restrict__ counts,
                             float* __restrict__ sum_mu) {
  int idx = blockIdx.x * 256 + threadIdx.x;  // over B*KD
  int b = idx >> 7, k = idx & 127;
  if (b < B_ROWS) {
    int l = labels[b];
    atomicAdd(&sum_mu[l * KD + k], z[idx]);
    if (k == 0) atomicAdd(&counts[l], 1.0f);
  }
}

// St = raw/B - mu mu^T + I
__global__ void sigtotal_ep(float* __restrict__ St, const float* __restrict__ mu) {
  int idx = blockIdx.x * 256 + threadIdx.x;  // 128*128
  int i = idx >> 7, j = idx & 127;
  St[idx] = St[idx] / (float)B_ROWS - mu[i] * mu[j] + ((i == j) ? 1.0f : 0.0f);
}

// SF[l] = (raw_l - mu s_l^T - s_l mu^T + c_l mu mu^T)/safe_c + I
__global__ void sigF_ep(float* __restrict__ SF, const float* __restrict__ counts,
                        const float* __restrict__ mu,
                        const float* __restrict__ smu) {
  int idx = blockIdx.x * 256 + threadIdx.x;  // 10*128*128
  int l = idx >> 14, r = idx & 16383;
  int i = r >> 7, j = r & 127;
  float c = counts[l];
  float sc = (c > 0.0f) ? c : 1.0f;
  float v = SF[idx] - mu[i] * smu[l * KD + j] - smu[l * KD + i] * mu[j] +
            c * mu[i] * mu[j];
  SF[idx] = v / sc + ((i == j) ? 1.0f : 0.0f);
}

// -------------------------------------------------------------------------
// Cholesky (upper R, src = R^T R) of 128x128 SPD in dynamic LDS; optional
// logdet[blockIdx.x] and R^{-1} (dense, zeros below diag).
// -------------------------------------------------------------------------
__global__ __launch_bounds__(128) void chol_kernel(
    const float* __restrict__ src, long long srcStride,
    float* __restrict__ rinv, float* __restrict__ logdet) {
  extern __shared__ float sm[];
  float (*Am)[128] = (float(*)[128])sm;
  const int t = threadIdx.x;
  const float* S = src + (size_t)blockIdx.x * (size_t)srcStride;
  for (int i = 0; i < 128; ++i) Am[i][t] = S[i * 128 + t];
  __syncthreads();
  for (int j = 0; j < 128; ++j) {
    float d = sqrtf(fmaxf(Am[j][j], 1e-30f));
    __syncthreads();
    if (t >= j) Am[j][t] = Am[j][t] / d;
    __syncthreads();
    if (t > j) {
      float rjt = Am[j][t];
      for (int i = j + 1; i <= t; ++i) Am[i][t] -= Am[j][i] * rjt;
    }
    __syncthreads();
  }
  if (logdet && t == 0) {
    float s = 0.0f;
    for (int j = 0; j < 128; ++j) s += logf(Am[j][j]);
    logdet[blockIdx.x] = 2.0f * s;
  }
  if (rinv) {
    float xl[128];
    for (int i = 0; i < 128; ++i) xl[i] = 0.0f;
    xl[t] = 1.0f / Am[t][t];
    for (int i = t - 1; i >= 0; --i) {
      float s = 0.0f;
      for (int l = i + 1; l <= t; ++l) s += Am[i][l] * xl[l];
      xl[i] = -s / Am[i][i];
    }
    for (int i = 0; i < 128; ++i) rinv[i * 128 + t] = (i <= t) ? xl[i] : 0.0f;
  }
}

// Sinv = X * X^T, X = R^{-1} upper.
__global__ void syminv_kernel(const float* __restrict__ X,
                              float* __restrict__ Sinv) {
  int idx = blockIdx.x * 256 + threadIdx.x;  // 128*128
  int i = idx >> 7, j = idx & 127;
  int lo = (i > j) ? i : j;
  float s = 0.0f;
  for (int l = lo; l < 128; ++l) s += X[i * 128 + l] * X[j * 128 + l];
  Sinv[idx] = s;
}

// -------------------------------------------------------------------------
__global__ __launch_bounds__(128) void final_kernel(
    const float* __restrict__ Sinv, const float* __restrict__ SF,
    const float* __restrict__ counts, const float* __restrict__ sum_mu,
    const float* __restrict__ mu_total, const float* __restrict__ logdets,
    float* __restrict__ out) {
  __shared__ float red[128];
  __shared__ float dvec[128];
  const int t = threadIdx.x;
  float total = 0.0f, nk = 0.0f;
  const float logdet_t = logdets[0];
  for (int l = 0; l < L_LAB; ++l) {
    float c = counts[l];
    float sc = (c > 0.0f) ? c : 1.0f;
    dvec[t] = mu_total[t] - sum_mu[l * KD + t] / sc;
    __syncthreads();
    float p = 0.0f;
    for (int j = 0; j < 128; ++j)
      p += Sinv[t * 128 + j] * SF[l * 16384 + j * 128 + t];
    float e = 0.0f;
    for (int j = 0; j < 128; ++j) e += Sinv[t * 128 + j] * dvec[j];
    red[t] = p + dvec[t] * e;
    __syncthreads();
    for (int s = 64; s > 0; s >>= 1) {
      if (t < s) red[t] += red[t + s];
      __syncthreads();
    }
    if (t == 0) {
      float kl = 0.5f * (red[0] - (float)KD + logdet_t - logdets[1 + l]);
      if (c > 0.0f) { total += kl; nk += 1.0f; }
    }
    __syncthreads();
  }
  if (t == 0) out[0] = total / (nk * (float)B_ROWS);
}

// -------------------------------------------------------------------------
extern "C" void kernel_launch(void* const* d_in, const int* in_sizes, int n_in,
                              void* d_out, int out_size, void* d_ws,
                              size_t ws_size, hipStream_t stream) {
  (void)in_sizes; (void)n_in; (void)out_size; (void)ws_size;
  const float* x = (const float*)d_in[0];
  const int* labels = (const int*)d_in[1];
  float* out = (float*)d_out;

  float* ws      = (float*)d_ws;
  float* mean    = ws;                  // 1024
  float* C       = mean + 1024;         // 1024*1024
  float* Uk      = C + 1024 * 1024;     // 1024*128
  float* Y       = Uk + 1024 * 128;     // 1024*128
  float* G       = Y + 1024 * 128;      // 128*128
  float* Rinv    = G + 16384;           // 128*128
  float* z       = Rinv + 16384;        // 8192*128
  float* mu_t    = z + 8192 * 128;      // 128
  float* counts  = mu_t + 128;          // 16
  float* sum_mu  = counts + 16;         // 10*128
  float* St      = sum_mu + 1280;       // 128*128
  float* SF      = St + 16384;          // 10*128*128
  float* Sinv    = SF + 163840;         // 128*128
  float* logdets = Sinv + 16384;        // 11
  float* w       = logdets + 16;        // 128

  // 1) column means; raw Gram C = x^T x (WMMA, async-LDS staged); de-center.
  colmean_kernel<<<dim3(4), dim3(256), 0, stream>>>(x, B_ROWS, D_DIM, mean);
  gram_kernel<<<dim3(8, 16, 1), dim3(256), 0, stream>>>(x, nullptr, C, D_DIM,
                                                        B_ROWS);
  rank1_ep<<<dim3(4096), dim3(256), 0, stream>>>(C, mean);
  // 2) subspace iteration -> orthonormal basis of top-128 eigenspace of C.
  init_kernel<<<dim3((D_DIM * KD) / 256), dim3(256), 0, stream>>>(Uk,
                                                                  D_DIM * KD);
  for (int it = 0; it < POWER_ITERS; ++it) {
    gemm_kernel<<<dim3(8, 2), dim3(256), 0, stream>>>(C, D_DIM, Uk, Y, D_DIM);
    gram_kernel<<<dim3(1, 2, 1), dim3(256), 0, stream>>>(Y, nullptr, G, KD,
                                                         D_DIM);
    chol_kernel<<<dim3(1), dim3(128), 128 * 128 * 4, stream>>>(G, 0, Rinv,
                                                               nullptr);
    gemm_kernel<<<dim3(8, 2), dim3(256), 0, stream>>>(Y, KD, Rinv, Uk, KD);
  }
  // 3) z = x @ U, then shift rows by mean^T U  (== xc @ U).
  gemm_kernel<<<dim3(64, 2), dim3(256), 0, stream>>>(x, D_DIM, Uk, z, D_DIM);
  meanproj_kernel<<<dim3(1), dim3(128), 0, stream>>>(mean, Uk, w);
  zshift_kernel<<<dim3(4096), dim3(256), 0, stream>>>(z, w);
  // 4) mu_total, per-label counts / sums.
  colmean_kernel<<<dim3(1), dim3(128), 0, stream>>>(z, B_ROWS, KD, mu_t);
  zero_kernel<<<dim3(6), dim3(256), 0, stream>>>(counts, 16 + 1280);
  accum_kernel<<<dim3((B_ROWS * KD) / 256), dim3(256), 0, stream>>>(z, labels,
                                                                    counts,
                                                                    sum_mu);
  // 5) Sigma_total (raw WMMA gram + rank-1 epilogue).
  gram_kernel<<<dim3(1, 2, 1), dim3(256), 0, stream>>>(z, nullptr, St, KD,
                                                       B_ROWS);
  sigtotal_ep<<<dim3(64), dim3(256), 0, stream>>>(St, mu_t);
  // 6) Sigma_F: 10 label-masked raw WMMA grams + epilogue.
  gram_kernel<<<dim3(1, 2, L_LAB), dim3(256), 0, stream>>>(z, labels, SF, KD,
                                                           B_ROWS);
  sigF_ep<<<dim3(640), dim3(256), 0, stream>>>(SF, counts, mu_t, sum_mu);
  // 7) inverses + logdets.
  chol_kernel<<<dim3(1), dim3(128), 128 * 128 * 4, stream>>>(St, 0, Rinv,
                                                             &logdets[0]);
  syminv_kernel<<<dim3(64), dim3(256), 0, stream>>>(Rinv, Sinv);
  chol_kernel<<<dim3(L_LAB), dim3(128), 128 * 128 * 4, stream>>>(SF, 16384,
                                                                 nullptr,
                                                                 &logdets[1]);
  // 8) KL reduction -> scalar.
  final_kernel<<<dim3(1), dim3(128), 0, stream>>>(Sinv, SF, counts, sum_mu,
                                                  mu_t, logdets, out);
}